// MSCELoss_65970697667422
// MI455X (gfx1250) — compile-verified
//
#include <hip/hip_runtime.h>
#include <hip/hip_bf16.h>
#include <stdint.h>

typedef float v2f __attribute__((ext_vector_type(2)));
typedef float v8f __attribute__((ext_vector_type(8)));

static constexpr int IMG_H = 256;
static constexpr int IMG_W = 256;
static constexpr int NPIX  = IMG_H * IMG_W;

__constant__ float GK[5] = {0.054488685f, 0.244201347f, 0.402619946f,
                            0.244201347f, 0.054488685f};

__device__ __forceinline__ int reflect_idx(int i, int n) {
  if (i < 0) i = -i;
  if (i >= n) i = 2 * n - 2 - i;
  return i;
}

__device__ __forceinline__ float bandcoef(int d) {
  int dc = d < 0 ? 0 : (d > 4 ? 4 : d);
  float m = ((unsigned)d <= 4u) ? 1.0f : 0.0f;
  return m * GK[dc];
}

// ---------------------------------------------------------------------------
// Gaussian blur, horizontal pass via WMMA f32 16x16x4:
//   Out(16x16) = In(16x20) x BandedB(20x16), 5 K-chunks of 4.
// One wave per 16x16 tile. Reflect padding on columns.
// ---------------------------------------------------------------------------
__global__ void blurH_wmma(const float* __restrict__ in, float* __restrict__ out,
                           int nimg) {
  const int wave = blockIdx.x * (blockDim.x >> 5) + (threadIdx.x >> 5);
  const int lane = threadIdx.x & 31;
  const int img  = wave >> 8;            // 256 tiles per image
  if (img >= nimg) return;               // wave-uniform
  const int t  = wave & 255;
  const int r0 = (t >> 4) << 4;
  const int c0 = (t & 15) << 4;
  const float* src = in + (size_t)img * NPIX;
  float* dst       = out + (size_t)img * NPIX;

  const int half = lane >> 4;
  const int m    = lane & 15;            // A row / B-D column

  v8f acc = {};
#pragma unroll
  for (int j = 0; j < 5; ++j) {
    const int kb = 4 * j + 2 * half;     // K index of .x element (both A and B)
    v2f a, b;
    const int row = r0 + m;
    const int ca0 = reflect_idx(c0 - 2 + kb, IMG_W);
    const int ca1 = reflect_idx(c0 - 2 + kb + 1, IMG_W);
    a.x = src[row * IMG_W + ca0];
    a.y = src[row * IMG_W + ca1];
    b.x = bandcoef(kb - m);              // B[k][n] = g[k-n], n == m
    b.y = bandcoef(kb + 1 - m);
    acc = __builtin_amdgcn_wmma_f32_16x16x4_f32(false, a, false, b,
                                                (short)0, acc, false, false);
  }
#pragma unroll
  for (int r = 0; r < 8; ++r)
    dst[(r0 + r + 8 * half) * IMG_W + c0 + m] = acc[r];
}

// ---------------------------------------------------------------------------
// Gaussian blur, vertical pass via WMMA f32 16x16x4:
//   Out(16x16) = BandedA(16x20) x In(20x16). Reflect padding on rows.
// ---------------------------------------------------------------------------
__global__ void blurV_wmma(const float* __restrict__ in, float* __restrict__ out,
                           int nimg) {
  const int wave = blockIdx.x * (blockDim.x >> 5) + (threadIdx.x >> 5);
  const int lane = threadIdx.x & 31;
  const int img  = wave >> 8;
  if (img >= nimg) return;
  const int t  = wave & 255;
  const int r0 = (t >> 4) << 4;
  const int c0 = (t & 15) << 4;
  const float* src = in + (size_t)img * NPIX;
  float* dst       = out + (size_t)img * NPIX;

  const int half = lane >> 4;
  const int m    = lane & 15;

  v8f acc = {};
#pragma unroll
  for (int j = 0; j < 5; ++j) {
    const int kb = 4 * j + 2 * half;
    v2f a, b;
    a.x = bandcoef(kb - m);              // A[m][k] = g[k-m]
    a.y = bandcoef(kb + 1 - m);
    const int ra0 = reflect_idx(r0 - 2 + kb, IMG_H);
    const int ra1 = reflect_idx(r0 - 2 + kb + 1, IMG_H);
    b.x = src[ra0 * IMG_W + c0 + m];     // B[k][n], n == m
    b.y = src[ra1 * IMG_W + c0 + m];
    acc = __builtin_amdgcn_wmma_f32_16x16x4_f32(false, a, false, b,
                                                (short)0, acc, false, false);
  }
#pragma unroll
  for (int r = 0; r < 8; ++r)
    dst[(r0 + r + 8 * half) * IMG_W + c0 + m] = acc[r];
}

// ---------------------------------------------------------------------------
// Async global->LDS helper (CDNA5 ASYNCcnt path)
// ---------------------------------------------------------------------------
__device__ __forceinline__ void async_load_f32_to_lds(unsigned lds_byte_off,
                                                      const float* base,
                                                      unsigned byte_off) {
  asm volatile("global_load_async_to_lds_b32 %0, %1, %2"
               :
               : "v"(lds_byte_off), "v"(byte_off),
                 "s"((unsigned long long)(uintptr_t)base)
               : "memory");
}
__device__ __forceinline__ void wait_async_zero() {
  asm volatile("s_wait_asynccnt 0" ::: "memory");
}

// ---------------------------------------------------------------------------
// Sobel gradients + magnitude + quantized direction.
// 16x16 tile per block; 18x18 halo staged into LDS with async copies.
// 'edge' (clamp) padding per reference.
// ---------------------------------------------------------------------------
__global__ void sobel_mag_dir(const float* __restrict__ blur,
                              float* __restrict__ mag,
                              unsigned char* __restrict__ dir, int nimg) {
  __shared__ float sm[18 * 18];
  const int img = blockIdx.z;
  const int r0  = blockIdx.y * 16;
  const int c0  = blockIdx.x * 16;
  const float* src = blur + (size_t)img * NPIX;
  const int tid = threadIdx.y * 16 + threadIdx.x;

  for (int i = tid; i < 18 * 18; i += 256) {
    const int lr = i / 18, lc = i % 18;
    int gr = r0 + lr - 1; gr = gr < 0 ? 0 : (gr > IMG_H - 1 ? IMG_H - 1 : gr);
    int gc = c0 + lc - 1; gc = gc < 0 ? 0 : (gc > IMG_W - 1 ? IMG_W - 1 : gc);
    async_load_f32_to_lds((unsigned)(i * 4), src,
                          (unsigned)((gr * IMG_W + gc) * 4));
  }
  wait_async_zero();
  __syncthreads();

  const int lr = threadIdx.y + 1, lc = threadIdx.x + 1;
#define SM(dr, dc) sm[(lr + (dr)) * 18 + (lc + (dc))]
  const float gx = (SM(-1, 1) - SM(-1, -1)) + 2.0f * (SM(0, 1) - SM(0, -1)) +
                   (SM(1, 1) - SM(1, -1));
  const float gy = (SM(1, -1) - SM(-1, -1)) + 2.0f * (SM(1, 0) - SM(-1, 0)) +
                   (SM(1, 1) - SM(-1, 1));
#undef SM
  const float mg = sqrtf(gx * gx + gy * gy + 1e-6f);
  int q = __float2int_rn(atan2f(gy, gx) * 1.27323954473516f);  // rad -> /45deg
  q = ((q % 8) + 8) & 7;
  const size_t p =
      (size_t)img * NPIX + (size_t)(r0 + threadIdx.y) * IMG_W + c0 + threadIdx.x;
  mag[p] = mg;
  dir[p] = (unsigned char)q;
}

// ---------------------------------------------------------------------------
// NMS along quantized direction + double threshold -> class map {0,1,2}
// ---------------------------------------------------------------------------
__global__ void nms_thresh(const float* __restrict__ mag,
                           const unsigned char* __restrict__ dir,
                           unsigned char* __restrict__ e, int n) {
  const int i = blockIdx.x * blockDim.x + threadIdx.x;
  if (i >= n) return;
  const int pix = i & (NPIX - 1);
  const int r = pix >> 8, c = pix & 255;
  const int base = i - pix;
  const float mc = mag[i];
  const int d = dir[i];
  const int DR[8] = {0, -1, -1, -1, 0, 1, 1, 1};
  const int DC[8] = {1, 1, 0, -1, -1, -1, 0, 1};
  const int d2 = (d + 4) & 7;
  float npos = 0.0f, nneg = 0.0f;
  {
    const int rr = r + DR[d], cc = c + DC[d];
    if (rr >= 0 && rr < IMG_H && cc >= 0 && cc < IMG_W)
      npos = mag[base + rr * IMG_W + cc];
  }
  {
    const int rr = r + DR[d2], cc = c + DC[d2];
    if (rr >= 0 && rr < IMG_H && cc >= 0 && cc < IMG_W)
      nneg = mag[base + rr * IMG_W + cc];
  }
  const float pos = mc - npos, neg = mc - nneg;
  const float keep = (fminf(pos, neg) > 0.0f) ? mc : 0.0f;
  e[i] = (unsigned char)((keep > 0.1f) + (keep > 0.2f));
}

// ---------------------------------------------------------------------------
// Hysteresis promote: weak(1) adjacent (3x3) to strong(2) becomes strong.
// finalpass: emit binary {0,1} (weak without strong neighbor drops to 0).
// ---------------------------------------------------------------------------
__global__ void promote(const unsigned char* __restrict__ in,
                        unsigned char* __restrict__ out, int n, int finalpass) {
  const int i = blockIdx.x * blockDim.x + threadIdx.x;
  if (i >= n) return;
  const int pix = i & (NPIX - 1);
  const int r = pix >> 8, c = pix & 255;
  const int base = i - pix;
  const unsigned char v = in[i];
  int strongN = 0;
  if (v == 1) {
#pragma unroll
    for (int dr = -1; dr <= 1; ++dr)
#pragma unroll
      for (int dc = -1; dc <= 1; ++dc) {
        const int rr = r + dr, cc = c + dc;
        if (rr >= 0 && rr < IMG_H && cc >= 0 && cc < IMG_W)
          strongN |= (in[base + rr * IMG_W + cc] == 2);
      }
  }
  unsigned char o;
  if (!finalpass)
    o = (v == 2) ? 2 : ((v == 1) ? (strongN ? 2 : 1) : 0);
  else
    o = (unsigned char)((v == 2) || (v == 1 && strongN));
  out[i] = o;
}

// ---------------------------------------------------------------------------
// Deterministic two-stage reduction of 0.85*mse(img) + 0.15*mse(edges)
// ---------------------------------------------------------------------------
__global__ void reduce_partial(const float* __restrict__ a,
                               const float* __restrict__ b,
                               const unsigned char* __restrict__ ea,
                               const unsigned char* __restrict__ eb,
                               float* __restrict__ partials, int n) {
  __shared__ float r1[256], r2[256];
  const int tid = threadIdx.x;
  float s1 = 0.0f, s2 = 0.0f;
  const int base = blockIdx.x * 2048 + tid;
#pragma unroll
  for (int k = 0; k < 8; ++k) {
    const int i = base + k * 256;
    if (i < n) {
      const float d = a[i] - b[i];
      s1 += d * d;
      s2 += (ea[i] != eb[i]) ? 1.0f : 0.0f;
    }
  }
  r1[tid] = s1;
  r2[tid] = s2;
  __syncthreads();
  for (int s = 128; s > 0; s >>= 1) {
    if (tid < s) {
      r1[tid] += r1[tid + s];
      r2[tid] += r2[tid + s];
    }
    __syncthreads();
  }
  if (tid == 0) {
    partials[blockIdx.x * 2 + 0] = r1[0];
    partials[blockIdx.x * 2 + 1] = r2[0];
  }
}

__global__ void reduce_final(const float* __restrict__ partials, int nb,
                             float* __restrict__ out, float invN) {
  __shared__ float r1[256], r2[256];
  const int tid = threadIdx.x;
  float s1 = 0.0f, s2 = 0.0f;
  for (int i = tid; i < nb; i += 256) {
    s1 += partials[2 * i + 0];
    s2 += partials[2 * i + 1];
  }
  r1[tid] = s1;
  r2[tid] = s2;
  __syncthreads();
  for (int s = 128; s > 0; s >>= 1) {
    if (tid < s) {
      r1[tid] += r1[tid + s];
      r2[tid] += r2[tid + s];
    }
    __syncthreads();
  }
  if (tid == 0) out[0] = 0.85f * r1[0] * invN + 0.15f * r2[0] * invN;
}

// ---------------------------------------------------------------------------
extern "C" void kernel_launch(void* const* d_in, const int* in_sizes, int n_in,
                              void* d_out, int out_size, void* d_ws,
                              size_t ws_size, hipStream_t stream) {
  (void)n_in; (void)out_size; (void)ws_size;
  const float* pred  = (const float*)d_in[0];
  const float* batch = (const float*)d_in[1];
  const int n    = in_sizes[0];
  const int nimg = n / NPIX;

  char* ws = (char*)d_ws;
  float* t0 = (float*)ws;                          // blurH out / mag
  float* t1 = (float*)(ws + (size_t)n * 4);        // blurred
  unsigned char* dirb = (unsigned char*)(ws + (size_t)n * 8);  // dir / pingpong
  unsigned char* work = dirb + (size_t)n;          // class map pingpong
  unsigned char* eA   = work + (size_t)n;          // final binary edges (pred)
  unsigned char* eB   = eA + (size_t)n;            // final binary edges (batch)
  float* partials = (float*)(eB + (size_t)n);

  const int waves      = nimg * 256;
  const int blurBlocks = waves / 4;                // 4 waves per 128-thread block
  const int pixBlocks  = (n + 255) / 256;
  const dim3 sg(16, 16, nimg), sb(16, 16);

  for (int which = 0; which < 2; ++which) {
    const float* img   = which ? batch : pred;
    unsigned char* eOut = which ? eB : eA;
    blurH_wmma<<<blurBlocks, 128, 0, stream>>>(img, t0, nimg);
    blurV_wmma<<<blurBlocks, 128, 0, stream>>>(t0, t1, nimg);
    sobel_mag_dir<<<sg, sb, 0, stream>>>(t1, t0, dirb, nimg);
    nms_thresh<<<pixBlocks, 256, 0, stream>>>(t0, dirb, work, n);
    unsigned char* src = work;
    unsigned char* dst = dirb;
    for (int p = 0; p < 31; ++p) {                 // bounded hysteresis
      promote<<<pixBlocks, 256, 0, stream>>>(src, dst, n, 0);
      unsigned char* t = src; src = dst; dst = t;
    }
    promote<<<pixBlocks, 256, 0, stream>>>(src, eOut, n, 1);
  }

  const int nb = (n + 2047) / 2048;
  reduce_partial<<<nb, 256, 0, stream>>>(pred, batch, eA, eB, partials, n);
  reduce_final<<<1, 256, 0, stream>>>(partials, nb, (float*)d_out,
                                      1.0f / (float)n);
}